// KANFNOBlock_67963562492191
// MI455X (gfx1250) — compile-verified
//
#include <hip/hip_runtime.h>
#include <math.h>

// ---------------------------------------------------------------------------
// KAN-FNO block for gfx1250 (MI455X).  All GEMMs via V_WMMA_F32_16X16X4_F32.
// ---------------------------------------------------------------------------

typedef float v2f __attribute__((ext_vector_type(2)));
typedef float v8f __attribute__((ext_vector_type(8)));

__device__ __forceinline__ v8f wmma_f32x4(v2f a, v2f b, v8f c) {
    // (neg_a, A, neg_b, B, c_mod, C, reuse_a, reuse_b)
    return __builtin_amdgcn_wmma_f32_16x16x4_f32(false, a, false, b, (short)0, c,
                                                 false, false);
}

__device__ __forceinline__ float siluf(float x) {
    return x / (1.0f + expf(-x));
}

__device__ __forceinline__ float geluf(float x) {
    return 0.5f * x * (1.0f + erff(x * 0.70710678118654752f));
}

// Cubic B-spline basis on the efficient-kan grid: g[t] = 0.4*t - 2.2, t=0..11.
__device__ __forceinline__ void bspline8(float x, float* __restrict__ out) {
    const float h  = 0.4f;
    const float g0 = -2.2f;
    float b[11];
#pragma unroll
    for (int j = 0; j < 11; ++j) {
        float gj = g0 + h * (float)j;
        b[j] = (x >= gj && x < gj + h) ? 1.0f : 0.0f;
    }
#pragma unroll
    for (int k = 1; k <= 3; ++k) {
        float inv = 1.0f / (h * (float)k);
#pragma unroll
        for (int j = 0; j < 11 - 1; ++j) {   // only first 11-k entries meaningful
            if (j < 11 - k) {
                float gj = g0 + h * (float)j;
                float left  = (x - gj) * inv;
                float right = (gj + h * (float)(k + 1) - x) * inv;
                b[j] = left * b[j] + right * b[j + 1];
            }
        }
    }
#pragma unroll
    for (int j = 0; j < 8; ++j) out[j] = b[j];
}

// ---------------------------------------------------------------------------
// Kernel 1: repack all weights into K-major [k][o] layouts for WMMA B-tiles.
// ---------------------------------------------------------------------------
__global__ void pack_weights(const float* __restrict__ w1r, const float* __restrict__ w1i,
                             const float* __restrict__ w2r, const float* __restrict__ w2i,
                             const float* __restrict__ convW,
                             const float* __restrict__ k1b, const float* __restrict__ k1s,
                             const float* __restrict__ k1sc,
                             const float* __restrict__ k2b, const float* __restrict__ k2s,
                             const float* __restrict__ k2sc,
                             float* __restrict__ WspR, float* __restrict__ WspI,
                             float* __restrict__ Wc,
                             float* __restrict__ Wb1, float* __restrict__ Wb2,
                             float* __restrict__ Ws1, float* __restrict__ Ws2) {
    const int NSPEC = 512 * 64 * 64;
    int gid = blockIdx.x * blockDim.x + threadIdx.x;
    if (gid < NSPEC) {
        int m = gid >> 12;             // mode 0..511: kyIdx = m>>4, kx = m&15
        int rem = gid & 4095;
        int i = rem >> 6, o = rem & 63;
        int kyIdx = m >> 4, kx = m & 15;
        int kyW = kyIdx & 15;
        int src = ((i * 64 + o) * 16 + kyW) * 16 + kx;   // (i,o,ky,kx)
        if (kyIdx < 16) { WspR[gid] = w1r[src]; WspI[gid] = w1i[src]; }
        else            { WspR[gid] = w2r[src]; WspI[gid] = w2i[src]; }
        return;
    }
    int g = gid - NSPEC;
    if (g < 4096) {                       // Wc[c][o] = convW[o][c]
        int c = g >> 6, o = g & 63;
        Wc[g] = convW[o * 64 + c];
    } else if (g < 8192) {                // Wb1[i][o] = k1_base[o][i]
        int g2 = g - 4096;
        int i = g2 >> 6, o = g2 & 63;
        Wb1[g2] = k1b[o * 64 + i];
    } else if (g < 12288) {
        int g2 = g - 8192;
        int i = g2 >> 6, o = g2 & 63;
        Wb2[g2] = k2b[o * 64 + i];
    } else if (g < 12288 + 32768) {       // Ws1[(i*8+j)][o] = spline[o][i][j]*scaler[o][i]
        int g2 = g - 12288;
        int k = g2 >> 6, o = g2 & 63;
        int i = k >> 3, j = k & 7;
        Ws1[g2] = k1s[(o * 64 + i) * 8 + j] * k1sc[o * 64 + i];
    } else if (g < 12288 + 65536) {
        int g2 = g - (12288 + 32768);
        int k = g2 >> 6, o = g2 & 63;
        int i = k >> 3, j = k & 7;
        Ws2[g2] = k2s[(o * 64 + i) * 8 + j] * k2sc[o * 64 + i];
    }
}

// ---------------------------------------------------------------------------
// Kernel 2: truncated forward DFT.  One block per (b,c) image.
// xft[mode][b][c], mode = kyIdx*16 + kx; kyIdx 0..15 -> rows 0..15 (w1),
// kyIdx 16..31 -> rows 112..127 (w2).
// ---------------------------------------------------------------------------
__global__ void dft_fwd(const float* __restrict__ x,
                        float* __restrict__ xftR, float* __restrict__ xftI) {
    extern __shared__ float smem[];
    float* img = smem;           // 16384
    float* twr = img + 16384;    // 128
    float* twi = twr + 128;      // 128
    float* Tr  = twi + 128;      // 4096
    float* Ti  = Tr + 4096;      // 4096

    int bc = blockIdx.x;
    int b = bc >> 6, c = bc & 63;
    const float* src = x + (size_t)(b * 64 + c) * 16384;
    int tid = threadIdx.x;

    if (tid < 128) {
        float s, co;
        sincosf(-6.283185307179586f * (float)tid / 128.0f, &s, &co);
        twr[tid] = co; twi[tid] = s;     // e^{-2*pi*i*t/128}
    }
    for (int i = tid; i < 4096; i += 256)
        ((float4*)img)[i] = ((const float4*)src)[i];
    __syncthreads();

    // Stage 1: T[kyIdx][w] = sum_h img[h][w] * e^{-2pi i kyA h / 128}
    for (int idx = tid; idx < 32 * 128; idx += 256) {
        int kyIdx = idx >> 7, w = idx & 127;
        int kyA = (kyIdx < 16) ? kyIdx : kyIdx + 96;
        float sr = 0.0f, si = 0.0f;
        for (int hh = 0; hh < 128; ++hh) {
            float v = img[hh * 128 + w];
            int tt = (kyA * hh) & 127;
            sr = fmaf(v, twr[tt], sr);
            si = fmaf(v, twi[tt], si);
        }
        Tr[idx] = sr; Ti[idx] = si;
    }
    __syncthreads();

    // Stage 2: X[kyIdx][kx] = sum_w T[kyIdx][w] * e^{-2pi i kx w / 128}
    for (int idx = tid; idx < 512; idx += 256) {
        int kyIdx = idx >> 4, kx = idx & 15;
        float sr = 0.0f, si = 0.0f;
        for (int w = 0; w < 128; ++w) {
            int tt = (kx * w) & 127;
            float ar = Tr[kyIdx * 128 + w], ai = Ti[kyIdx * 128 + w];
            float br = twr[tt], bi = twi[tt];
            sr += ar * br - ai * bi;
            si += ar * bi + ai * br;
        }
        xftR[(size_t)idx * 1024 + b * 64 + c] = sr;
        xftI[(size_t)idx * 1024 + b * 64 + c] = si;
    }
}

// ---------------------------------------------------------------------------
// Kernel 3: per-mode complex GEMM  out[b][o] = sum_i X[b][i] * W[i][o].
// 512 blocks, 128 threads (4 waves, one 16-col N-tile each).  M=16 (batch).
// ---------------------------------------------------------------------------
__global__ void spec_mm(const float* __restrict__ xftR, const float* __restrict__ xftI,
                        const float* __restrict__ WspR, const float* __restrict__ WspI,
                        float* __restrict__ oftR, float* __restrict__ oftI) {
    int m = blockIdx.x;
    const float* ar = xftR + (size_t)m * 1024;   // [b(16)][i(64)]
    const float* ai = xftI + (size_t)m * 1024;
    const float* wr = WspR + (size_t)m * 4096;   // [k(64)][o(64)]
    const float* wi = WspI + (size_t)m * 4096;

    int lane = threadIdx.x & 31, wave = threadIdx.x >> 5;
    int n0 = wave * 16;
    int lm = lane & 15;
    int lk = (lane >> 4) * 2;

    v8f cr = {0, 0, 0, 0, 0, 0, 0, 0};
    v8f ci = {0, 0, 0, 0, 0, 0, 0, 0};

#pragma unroll 4
    for (int k0 = 0; k0 < 64; k0 += 4) {
        int kk = k0 + lk;
        v2f Ar, Ai, Br, Bi;
        Ar.x = ar[lm * 64 + kk];     Ar.y = ar[lm * 64 + kk + 1];
        Ai.x = ai[lm * 64 + kk];     Ai.y = ai[lm * 64 + kk + 1];
        Br.x = wr[kk * 64 + n0 + lm]; Br.y = wr[(kk + 1) * 64 + n0 + lm];
        Bi.x = wi[kk * 64 + n0 + lm]; Bi.y = wi[(kk + 1) * 64 + n0 + lm];
        v2f nAi = -Ai;               // f32 WMMA has no A-neg modifier (ISA: F32 NEG = CNeg only)
        cr = wmma_f32x4(Ar,  Br, cr);
        cr = wmma_f32x4(nAi, Bi, cr);
        ci = wmma_f32x4(Ar,  Bi, ci);
        ci = wmma_f32x4(Ai,  Br, ci);
    }
#pragma unroll
    for (int r = 0; r < 8; ++r) {
        int mm = r + ((lane >= 16) ? 8 : 0);
        oftR[(size_t)m * 1024 + mm * 64 + n0 + lm] = cr[r];
        oftI[(size_t)m * 1024 + mm * 64 + n0 + lm] = ci[r];
    }
}

// ---------------------------------------------------------------------------
// Kernel 4: sparse-mode irfft2.  One block per (b,c); writes NHWC yBuf.
// ---------------------------------------------------------------------------
__global__ void dft_inv(const float* __restrict__ oftR, const float* __restrict__ oftI,
                        float* __restrict__ yBuf) {
    extern __shared__ float smem[];
    float* Fr  = smem;          // 512
    float* Fi  = Fr + 512;      // 512
    float* twr = Fi + 512;      // 128
    float* twi = twr + 128;     // 128
    float* Vr  = twi + 128;     // 2048
    float* Vi  = Vr + 2048;     // 2048

    int bc = blockIdx.x;
    int b = bc >> 6, c = bc & 63;
    int tid = threadIdx.x;

    if (tid < 128) {
        float s, co;
        sincosf(6.283185307179586f * (float)tid / 128.0f, &s, &co);
        twr[tid] = co; twi[tid] = s;     // e^{+2*pi*i*t/128}
    }
    for (int m = tid; m < 512; m += 256) {
        Fr[m] = oftR[(size_t)m * 1024 + b * 64 + c];
        Fi[m] = oftI[(size_t)m * 1024 + b * 64 + c];
    }
    __syncthreads();

    // Height inverse: V[h][kx] = sum_{kyIdx} F[kyIdx][kx] * e^{+2pi i kyA h / 128}
    for (int idx = tid; idx < 128 * 16; idx += 256) {
        int hh = idx >> 4, kx = idx & 15;
        float sr = 0.0f, si = 0.0f;
#pragma unroll
        for (int kyIdx = 0; kyIdx < 32; ++kyIdx) {
            int kyA = (kyIdx < 16) ? kyIdx : kyIdx + 96;
            int tt = (kyA * hh) & 127;
            float fr = Fr[kyIdx * 16 + kx], fi = Fi[kyIdx * 16 + kx];
            float br = twr[tt], bi = twi[tt];
            sr += fr * br - fi * bi;
            si += fr * bi + fi * br;
        }
        Vr[idx] = sr; Vi[idx] = si;
    }
    __syncthreads();

    // Width irfft with hermitian doubling: y = (Re(V0) + 2*sum Re(Vk e^{i th})) / (H*W)
    const float inv = 1.0f / 16384.0f;
    for (int idx = tid; idx < 16384; idx += 256) {
        int hh = idx >> 7, w = idx & 127;
        float s = Vr[hh * 16];
#pragma unroll
        for (int kx = 1; kx < 16; ++kx) {
            int tt = (kx * w) & 127;
            s += 2.0f * (Vr[hh * 16 + kx] * twr[tt] - Vi[hh * 16 + kx] * twi[tt]);
        }
        yBuf[(size_t)(b * 16384 + idx) * 64 + c] = s * inv;
    }
}

// ---------------------------------------------------------------------------
// Kernel 5: fused 1x1 conv + bias + residual: yBuf += x @ Wc + bias (WMMA).
// 64 rows per block, 4 waves, N=64.
// ---------------------------------------------------------------------------
__global__ void conv1x1(const float* __restrict__ x, float* __restrict__ yBuf,
                        const float* __restrict__ Wc, const float* __restrict__ bias) {
    int rows0 = blockIdx.x * 64;
    int lane = threadIdx.x & 31, wave = threadIdx.x >> 5;
    int mBase = wave * 16;
    int lm = lane & 15;
    int lk = (lane >> 4) * 2;

    int b = rows0 >> 14;                       // 16384 pixels per batch
    int hwBase = rows0 & 16383;
    const float* xb = x + (size_t)b * 64 * 16384;
    int hw = hwBase + mBase + lm;              // A-tile row -> pixel

    for (int n0 = 0; n0 < 64; n0 += 16) {
        v8f acc;
#pragma unroll
        for (int r = 0; r < 8; ++r) {
            int mm = r + ((lane >= 16) ? 8 : 0);
            acc[r] = yBuf[(size_t)(rows0 + mBase + mm) * 64 + n0 + lm] + bias[n0 + lm];
        }
#pragma unroll 4
        for (int k0 = 0; k0 < 64; k0 += 4) {
            int kk = k0 + lk;
            v2f a, bv;
            a.x  = xb[(size_t)kk * 16384 + hw];
            a.y  = xb[(size_t)(kk + 1) * 16384 + hw];
            bv.x = Wc[kk * 64 + n0 + lm];
            bv.y = Wc[(kk + 1) * 64 + n0 + lm];
            acc = wmma_f32x4(a, bv, acc);
        }
#pragma unroll
        for (int r = 0; r < 8; ++r) {
            int mm = r + ((lane >= 16) ? 8 : 0);
            yBuf[(size_t)(rows0 + mBase + mm) * 64 + n0 + lm] = acc[r];
        }
    }
}

// ---------------------------------------------------------------------------
// Kernels 6/7: KAN layer.  out = silu(y) @ Wb  +  Bspline(y) @ Ws
// 64 rows/block, 4 waves.  LDS: basis 64x(512+4), silu 64x(64+4),
// FINAL adds a 64x(64+4) staging tile for coalesced NCHW stores + GELU.
// ---------------------------------------------------------------------------
template <bool FINAL>
__global__ void kan_layer(const float* __restrict__ yIn, float* __restrict__ yOut,
                          const float* __restrict__ Wb, const float* __restrict__ Ws) {
    extern __shared__ float smem[];
    const int BSS = 516;   // padded stride for basis (bank-conflict-free)
    const int ZSS = 68;    // padded stride for silu tile
    float* bS = smem;                  // 64 * 516
    float* zS = bS + 64 * BSS;         // 64 * 68
    float* oS = zS + 64 * ZSS;         // 64 * 68 (FINAL only)

    int rows0 = blockIdx.x * 64;
    int tid = threadIdx.x;

    // ---- Phase A: silu + B-spline basis into LDS ----
    for (int idx = tid; idx < 64 * 64; idx += 128) {
        int r = idx >> 6, c = idx & 63;
        float v = yIn[(size_t)(rows0 + r) * 64 + c];
        zS[r * ZSS + c] = siluf(v);
        float bb[8];
        bspline8(v, bb);
#pragma unroll
        for (int j = 0; j < 8; ++j) bS[r * BSS + c * 8 + j] = bb[j];
    }
    __syncthreads();

    // ---- Phase B: WMMA chains ----
    int lane = tid & 31, wave = tid >> 5;
    int mBase = wave * 16;
    int lm = lane & 15;
    int lk = (lane >> 4) * 2;
    const float* zRow = zS + (mBase + lm) * ZSS;
    const float* bRow = bS + (mBase + lm) * BSS;

    for (int n0 = 0; n0 < 64; n0 += 16) {
        v8f acc = {0, 0, 0, 0, 0, 0, 0, 0};
        // base branch, K = 64
#pragma unroll 4
        for (int k0 = 0; k0 < 64; k0 += 4) {
            int kk = k0 + lk;
            v2f a = *(const v2f*)(zRow + kk);                    // ds_load_b64
            v2f b;
            b.x = Wb[kk * 64 + n0 + lm];
            b.y = Wb[(kk + 1) * 64 + n0 + lm];
            acc = wmma_f32x4(a, b, acc);
        }
        // spline branch, K = 512
#pragma unroll 4
        for (int k0 = 0; k0 < 512; k0 += 4) {
            int kk = k0 + lk;
            v2f a = *(const v2f*)(bRow + kk);
            v2f b;
            b.x = Ws[kk * 64 + n0 + lm];
            b.y = Ws[(kk + 1) * 64 + n0 + lm];
            acc = wmma_f32x4(a, b, acc);
        }
        if (!FINAL) {
#pragma unroll
            for (int r = 0; r < 8; ++r) {
                int mm = r + ((lane >= 16) ? 8 : 0);
                yOut[(size_t)(rows0 + mBase + mm) * 64 + n0 + lm] = acc[r];
            }
        } else {
#pragma unroll
            for (int r = 0; r < 8; ++r) {
                int mm = r + ((lane >= 16) ? 8 : 0);
                oS[(mBase + mm) * ZSS + n0 + lm] = geluf(acc[r]);
            }
        }
    }

    if (FINAL) {
        __syncthreads();
        // Coalesced NHWC -> NCHW store: 64 consecutive pixels per channel.
        int b = rows0 >> 14;
        int hw0 = rows0 & 16383;
        for (int idx = tid; idx < 64 * 64; idx += 128) {
            int o = idx >> 6, r = idx & 63;
            yOut[(size_t)(b * 64 + o) * 16384 + hw0 + r] = oS[r * ZSS + o];
        }
    }
}

// ---------------------------------------------------------------------------
// Host-side launch
// ---------------------------------------------------------------------------
extern "C" void kernel_launch(void* const* d_in, const int* in_sizes, int n_in,
                              void* d_out, int out_size, void* d_ws, size_t ws_size,
                              hipStream_t stream) {
    const float* x      = (const float*)d_in[0];
    const float* w1r    = (const float*)d_in[1];
    const float* w1i    = (const float*)d_in[2];
    const float* w2r    = (const float*)d_in[3];
    const float* w2i    = (const float*)d_in[4];
    const float* convW  = (const float*)d_in[5];
    const float* convB  = (const float*)d_in[6];
    const float* k1b    = (const float*)d_in[7];
    const float* k1s    = (const float*)d_in[8];
    const float* k1sc   = (const float*)d_in[9];
    const float* k2b    = (const float*)d_in[10];
    const float* k2s    = (const float*)d_in[11];
    const float* k2sc   = (const float*)d_in[12];
    float* out = (float*)d_out;

    // Workspace layout (floats)
    float* yBuf = (float*)d_ws;              // 16,777,216  (NHWC y)
    float* xftR = yBuf + 16777216;           // 524,288
    float* xftI = xftR + 524288;
    float* oftR = xftI + 524288;
    float* oftI = oftR + 524288;
    float* WspR = oftI + 524288;             // 2,097,152
    float* WspI = WspR + 2097152;
    float* Wc   = WspI + 2097152;            // 4,096
    float* Wb1  = Wc + 4096;
    float* Wb2  = Wb1 + 4096;
    float* Ws1  = Wb2 + 4096;                // 32,768
    float* Ws2  = Ws1 + 32768;               // 32,768

    // 1. Weight repack
    {
        int total = 512 * 64 * 64 + 12288 + 65536;
        int blocks = (total + 255) / 256;
        pack_weights<<<blocks, 256, 0, stream>>>(w1r, w1i, w2r, w2i, convW,
                                                 k1b, k1s, k1sc, k2b, k2s, k2sc,
                                                 WspR, WspI, Wc, Wb1, Wb2, Ws1, Ws2);
    }
    // 2. Truncated forward DFT (one block per (b,c) image)
    dft_fwd<<<1024, 256, (16384 + 256 + 8192) * sizeof(float), stream>>>(x, xftR, xftI);
    // 3. Per-mode complex GEMM
    spec_mm<<<512, 128, 0, stream>>>(xftR, xftI, WspR, WspI, oftR, oftI);
    // 4. Sparse-mode inverse DFT -> NHWC yBuf
    dft_inv<<<1024, 256, (1024 + 256 + 4096) * sizeof(float), stream>>>(oftR, oftI, yBuf);
    // 5. Fused 1x1 conv + bias + residual (in place on yBuf)
    conv1x1<<<4096, 128, 0, stream>>>(x, yBuf, Wc, convB);
    // 6. KAN layer 1 (in place on yBuf)
    {
        size_t lds = (size_t)(64 * 516 + 64 * 68) * sizeof(float);
        kan_layer<false><<<4096, 128, lds, stream>>>(yBuf, yBuf, Wb1, Ws1);
    }
    // 7. KAN layer 2 + GELU + NCHW transpose -> d_out
    {
        size_t lds = (size_t)(64 * 516 + 2 * 64 * 68) * sizeof(float);
        kan_layer<true><<<4096, 128, lds, stream>>>(yBuf, out, Wb2, Ws2);
    }
}